// _VecPool2d_90288802496661
// MI455X (gfx1250) — compile-verified
//
#include <hip/hip_runtime.h>

#define B_  8
#define C_  256
#define H_  128
#define W_  128
#define HW_ (H_ * W_)
#define OH_ 65
#define OW_ 65

typedef __attribute__((ext_vector_type(2))) float v2f;
typedef __attribute__((ext_vector_type(8))) float v8f;

// ---------------------------------------------------------------------------
// Kernel 1: n[b,h,w] = sqrt(sum_c x[b,c,h,w]^2) via V_WMMA_F32_16X16X4_F32.
// Each wave handles 16 consecutive spatial positions (M dim), reduces 256
// channels in K=4 chunks. A[m][k] = x^2, B = all-ones => D[m][*] = row sums.
// A layout (ISA 16x4 f32): lanes 0-15 supply M=lane, K={0,1}; lanes 16-31
// supply M=lane-16, K={2,3}.  D col 0: lane 0 holds M=0..7, lane 16 M=8..15.
// ---------------------------------------------------------------------------
__global__ __launch_bounds__(256) void vecpool_norm_wmma(
    const float* __restrict__ x, float* __restrict__ nout) {
  const int lane = threadIdx.x & 31;
  const int wid  = (blockIdx.x * blockDim.x + threadIdx.x) >> 5;  // 0..8191
  const int b    = wid >> 10;          // 1024 waves per batch image
  const int hw0  = (wid & 1023) << 4;  // 16 positions per wave
  const int m    = lane & 15;
  const int k0   = (lane >> 4) << 1;   // 0 or 2

  const float* xp = x + (size_t)b * C_ * HW_ + hw0 + m;

  v2f bones = {1.0f, 1.0f};
  v8f acc = {};
#pragma unroll 8
  for (int c = 0; c < C_; c += 4) {
    float a0 = xp[(size_t)(c + k0) * HW_];
    float a1 = xp[(size_t)(c + k0 + 1) * HW_];
    v2f a = {a0 * a0, a1 * a1};
    acc = __builtin_amdgcn_wmma_f32_16x16x4_f32(
        /*neg_a=*/false, a, /*neg_b=*/false, bones,
        /*c_mod=*/(short)0, acc, /*reuse_a=*/false, /*reuse_b=*/false);
  }

  float* np = nout + (size_t)b * HW_ + hw0;
  if (lane == 0) {
#pragma unroll
    for (int i = 0; i < 8; ++i) np[i] = sqrtf(acc[i]);
  } else if (lane == 16) {
#pragma unroll
    for (int i = 0; i < 8; ++i) np[8 + i] = sqrtf(acc[i]);
  }
}

// ---------------------------------------------------------------------------
// Kernel 2: softmax-weighted 3x3/stride-2 pooling.
// Block = (oh, b, channel-quarter). Stages 3 norm rows (+zero halo) into LDS
// with async global->LDS loads, computes 9 softmax weights per ow once, then
// all threads sweep (c, ow) pairs doing the 9-tap weighted sum.
// ---------------------------------------------------------------------------
__global__ __launch_bounds__(128) void vecpool_pool(
    const float* __restrict__ x, const float* __restrict__ nrm,
    float* __restrict__ out) {
  __shared__ float s_n[3][136];   // padded norm rows: index = w + 2, w in [-2,130]
  __shared__ float s_w[9][OW_];   // softmax weights per window tap

  const int t  = threadIdx.x;                 // 0..127
  const int oh = blockIdx.x;                  // 0..64
  const int b  = blockIdx.y;                  // 0..7
  const int cb = blockIdx.z * (C_ / 4);       // channel base for this block

  // Zero the halo columns (w = -2,-1,128,129,130 and alignment pad).
  if (t < 24) {
    int r = t >> 3, i = t & 7;
    int col = (i < 2) ? i : (128 + i);        // 0,1,130..135
    s_n[r][col] = 0.0f;
  }

  const float* nb = nrm + (size_t)b * HW_;
#pragma unroll
  for (int r = 0; r < 3; ++r) {
    int ih = 2 * oh - 2 + r;
    if (ih >= 0 && ih < H_) {
      unsigned lds = (unsigned)(size_t)(void*)&s_n[r][2 + t];
      int off = (ih * W_ + t) * 4;
      asm volatile("global_load_async_to_lds_b32 %0, %1, %2"
                   :: "v"(lds), "v"(off), "s"(nb)
                   : "memory");
    } else {
      s_n[r][2 + t] = 0.0f;                   // fully padded row
    }
  }
  asm volatile("s_wait_asynccnt 0x0" ::: "memory");
  __syncthreads();

  // Softmax over the 9 window taps (padding contributes exp(0), as in ref).
  if (t < OW_) {
    float v[9];
#pragma unroll
    for (int kh = 0; kh < 3; ++kh)
#pragma unroll
      for (int kw = 0; kw < 3; ++kw)
        v[kh * 3 + kw] = s_n[kh][2 * t + kw]; // w = 2*t-2+kw  -> idx 2*t+kw
    float mx = v[0];
#pragma unroll
    for (int j = 1; j < 9; ++j) mx = fmaxf(mx, v[j]);
    float sum = 0.0f;
#pragma unroll
    for (int j = 0; j < 9; ++j) { v[j] = __expf(v[j] - mx); sum += v[j]; }
    float inv = 1.0f / sum;
#pragma unroll
    for (int j = 0; j < 9; ++j) s_w[j][t] = v[j] * inv;
  }
  __syncthreads();

  // Weighted 9-tap sum for 64 channels x 65 ow positions.
  const float* xb0 = x + ((size_t)(b * C_ + cb)) * HW_;
  float* ob0 = out + ((size_t)(b * C_ + cb) * OH_ + oh) * OW_;
  for (int idx = t; idx < (C_ / 4) * OW_; idx += 128) {
    int c  = idx / OW_;
    int ow = idx - c * OW_;
    const float* xc = xb0 + (size_t)c * HW_;
    float acc = 0.0f;
#pragma unroll
    for (int kh = 0; kh < 3; ++kh) {
      int ih = 2 * oh - 2 + kh;
      if (ih < 0 || ih >= H_) continue;       // uniform per block
      const float* xr = xc + ih * W_;
#pragma unroll
      for (int kw = 0; kw < 3; ++kw) {
        int iw = 2 * ow - 2 + kw;
        if (iw >= 0 && iw < W_)
          acc = fmaf(s_w[kh * 3 + kw][ow], xr[iw], acc);
      }
    }
    ob0[(size_t)c * (OH_ * OW_) + ow] = acc;
  }
}

extern "C" void kernel_launch(void* const* d_in, const int* in_sizes, int n_in,
                              void* d_out, int out_size, void* d_ws, size_t ws_size,
                              hipStream_t stream) {
  const float* x = (const float*)d_in[0];     // (8,256,128,128) fp32
  float* out = (float*)d_out;                 // (8,256,65,65)  fp32
  float* nrm = (float*)d_ws;                  // 8*128*128 floats = 512 KB scratch

  // Pass 1: channel L2 norms (8192 waves of 16 positions each).
  vecpool_norm_wmma<<<dim3(1024), dim3(256), 0, stream>>>(x, nrm);
  // Pass 2: softmax-weighted pooling; x is L2-resident from pass 1.
  vecpool_pool<<<dim3(OH_, B_, 4), dim3(128), 0, stream>>>(x, nrm, out);
}